// GNNLayer_6545530159666
// MI455X (gfx1250) — compile-verified
//
#include <hip/hip_runtime.h>
#include <math.h>

// ---------------------------------------------------------------------------
// GNN relational-attention layer for MI455X (gfx1250, wave32).
// Hoist hs@Ws (per-node) and hr@Wr (per-relation) out of the 600K-edge loop;
// edge stage is attn-combine + sigmoid + atomic scatter (all L2-resident).
// Big GEMMs (hidden@Ws, agg@Wh) run on V_WMMA_F32_16X16X4_F32 (exact f32):
// B staged in LDS once per block, one wave per 16-row x N strip, A fragment
// loaded once per k-step and reused across TN WMMA tiles, B fragments read
// from LDS into a register array so WMMAs issue in bursts.
// ---------------------------------------------------------------------------

typedef __attribute__((ext_vector_type(2))) float v2f;
typedef __attribute__((ext_vector_type(8))) float v8f;

#define N_NODE_C   50000
#define N_EDGE_C   600000
#define N_REL_C    474
#define IN_DIM_C   128
#define OUT_DIM_C  128
#define ATTN_DIM_C 64

// -------------------------------- zero fill --------------------------------
__global__ void gnn_zero_kernel(float* __restrict__ p, int n) {
    int i = blockIdx.x * blockDim.x + threadIdx.x;
    if (i < n) p[i] = 0.0f;
}

// ------------------- qr_vec = rela[q_rel[r_idx]] @ Wqr_w + b ---------------
__global__ void gnn_qr_kernel(const int* __restrict__ q_rel,
                              const int* __restrict__ r_idx,
                              const float* __restrict__ rela,
                              const float* __restrict__ Wqr_w,
                              const float* __restrict__ Wqr_b,
                              float* __restrict__ qr_vec) {
    int t = threadIdx.x;                      // 64 threads
    int r = q_rel[r_idx[0]];
    float s = Wqr_b[t];
    const float* hq = rela + (long long)r * IN_DIM_C;
    #pragma unroll 4
    for (int k = 0; k < IN_DIM_C; ++k)
        s += hq[k] * Wqr_w[k * ATTN_DIM_C + t];
    qr_vec[t] = s;
}

// ---------------- att_rel[r] = rela[r] @ Wr + qr_vec (474x64) --------------
__global__ void gnn_relatt_kernel(const float* __restrict__ rela,
                                  const float* __restrict__ Wr,
                                  const float* __restrict__ qr_vec,
                                  float* __restrict__ att_rel) {
    int r = blockIdx.x;                       // 474 blocks
    int t = threadIdx.x;                      // 64 threads
    float s = qr_vec[t];
    const float* hr = rela + (long long)r * IN_DIM_C;
    #pragma unroll 4
    for (int k = 0; k < IN_DIM_C; ++k)
        s += hr[k] * Wr[k * ATTN_DIM_C + t];
    att_rel[r * ATTN_DIM_C + t] = s;
}

// --------------------- f32 WMMA GEMM: C[M][NDIM] = A[M][128] @ B[128][NDIM]
// Block = 256 threads = 8 waves. B (128 x NDIM) is staged in LDS once per
// block (32/64 KB, fits the 320 KB WGP LDS). Wave w computes the full
// 16 x NDIM strip for M-tile (blockIdx*8 + w): per k-step one A fragment
// (global, b64) is reused by TN chained V_WMMA_F32_16X16X4_F32 whose B
// fragments come from LDS (separate DScnt, overlaps with global LOADcnt).
// Requirements (met here): M % 16 == 0, NDIM == 16*TN, K == 128.
// Fragment layouts per CDNA5 ISA 7.12.2:
//   A 16x4 (2 VGPR): lane<16 -> row=lane, K pair {0,1}; lane>=16 -> K {2,3}
//   B 4x16 (2 VGPR): lane<16 -> col=lane, K pair {0,1}; lane>=16 -> K {2,3}
//   C 16x16 (8 VGPR): col=lane&15, row = i + 8*(lane>>4)
template <int NDIM, int TN>
__global__ __launch_bounds__(256)
void gnn_wmma_gemm_f32(const float* __restrict__ A,
                       const float* __restrict__ B,
                       float* __restrict__ C,
                       int M) {
    __shared__ float Bs[IN_DIM_C * NDIM];

    // Cooperative stage of all of B into LDS (float4 per thread).
    {
        const float4* Bv = (const float4*)B;
        float4*       Sv = (float4*)Bs;
        const int nv = (IN_DIM_C * NDIM) / 4;
        for (int i = threadIdx.x; i < nv; i += 256)
            Sv[i] = Bv[i];
    }
    __syncthreads();

    const int lane = threadIdx.x & 31;
    const int wid  = threadIdx.x >> 5;
    const int tm   = blockIdx.x * 8 + wid;               // wave-uniform M tile
    if (tm >= (M >> 4)) return;                          // uniform exit: EXEC stays full

    const int half = lane >> 4;         // 0: lanes 0-15, 1: lanes 16-31
    const int l15  = lane & 15;

    const float* Arow = A + (long long)(tm * 16 + l15) * IN_DIM_C + 2 * half;

    v8f c[TN] = {};
    #pragma unroll 4
    for (int k = 0; k < IN_DIM_C; k += 4) {
        const v2f a = *(const v2f*)(Arow + k);           // 8B-aligned global b64
        const int kk = k + 2 * half;
        const float* B0 = Bs + kk * NDIM + l15;
        const float* B1 = B0 + NDIM;
        v2f b[TN];
        #pragma unroll
        for (int t = 0; t < TN; ++t) {                   // LDS load clause
            b[t].x = B0[t * 16];
            b[t].y = B1[t * 16];
        }
        #pragma unroll
        for (int t = 0; t < TN; ++t)                     // WMMA burst
            c[t] = __builtin_amdgcn_wmma_f32_16x16x4_f32(
                       /*neg_a=*/false, a, /*neg_b=*/false, b[t],
                       /*c_mod=*/(short)0, c[t], /*reuse_a=*/false, /*reuse_b=*/false);
    }

    const int rowBase = tm * 16 + 8 * half;
    #pragma unroll
    for (int t = 0; t < TN; ++t) {
        float* Cout = C + (long long)rowBase * NDIM + t * 16 + l15;
        #pragma unroll
        for (int i = 0; i < 8; ++i)
            Cout[(long long)i * NDIM] = c[t][i];
    }
}

// ------------------------------ edge stage ---------------------------------
// One wave32 per edge. Lanes split the 64 attention dims (2 each), tree-
// reduce the wa dot product with __shfl_xor, then each lane scatters 4 of
// the 128 message components with global_atomic_add_f32.
__global__ void gnn_edge_kernel(const int* __restrict__ edges,
                                const float* __restrict__ att_node,
                                const float* __restrict__ att_rel,
                                const float* __restrict__ hidden,
                                const float* __restrict__ rela,
                                const float* __restrict__ wa_w,
                                const float* __restrict__ wa_b,
                                float* __restrict__ agg) {
    const int lane = threadIdx.x & 31;
    const int wid  = threadIdx.x >> 5;
    const long long e = (long long)blockIdx.x * (blockDim.x >> 5) + wid;
    if (e >= N_EDGE_C) return;                 // wave-uniform

    const int sub = edges[e * 3 + 0];
    const int rel = edges[e * 3 + 1];
    const int obj = edges[e * 3 + 2];

    const float* an = att_node + (long long)sub * ATTN_DIM_C;
    const float* ar = att_rel  + (long long)rel * ATTN_DIM_C;

    const float p0 = an[lane]      + ar[lane];
    const float p1 = an[lane + 32] + ar[lane + 32];
    float s = fmaxf(p0, 0.0f) * wa_w[lane]
            + fmaxf(p1, 0.0f) * wa_w[lane + 32];

    #pragma unroll
    for (int off = 16; off > 0; off >>= 1)
        s += __shfl_xor(s, off, 32);

    const float alpha = 1.0f / (1.0f + __expf(-(s + wa_b[0])));

    const float* hs = hidden + (long long)sub * IN_DIM_C;
    const float* hr = rela   + (long long)rel * IN_DIM_C;
    float*       ao = agg    + (long long)obj * IN_DIM_C;
    #pragma unroll
    for (int i = 0; i < 4; ++i) {
        const int d = lane + 32 * i;
        atomicAdd(&ao[d], alpha * hs[d] * hr[d]);
    }
}

// ------------------------------- launcher ----------------------------------
extern "C" void kernel_launch(void* const* d_in, const int* in_sizes, int n_in,
                              void* d_out, int out_size, void* d_ws, size_t ws_size,
                              hipStream_t stream) {
    // Input order (setup_inputs): q_sub, q_rel, r_idx, hidden, edges, n_node,
    //                             rela_embed, Ws, Wr, Wqr_w, Wqr_b, wa_w, wa_b, Wh
    const int*   q_rel  = (const int*)  d_in[1];
    const int*   r_idx  = (const int*)  d_in[2];
    const float* hidden = (const float*)d_in[3];
    const int*   edges  = (const int*)  d_in[4];
    const float* rela   = (const float*)d_in[6];
    const float* Ws     = (const float*)d_in[7];
    const float* Wr     = (const float*)d_in[8];
    const float* Wqr_w  = (const float*)d_in[9];
    const float* Wqr_b  = (const float*)d_in[10];
    const float* wa_w   = (const float*)d_in[11];
    const float* wa_b   = (const float*)d_in[12];
    const float* Wh     = (const float*)d_in[13];
    float*       out    = (float*)d_out;

    // Workspace carve-up (all 256B aligned): ~38.6 MB total.
    char* ws = (char*)d_ws;
    float* att_node = (float*)ws;                                  // 50000*64
    ws += (size_t)N_NODE_C * ATTN_DIM_C * sizeof(float);
    float* att_rel  = (float*)ws;                                  // 474*64
    ws += ((size_t)N_REL_C * ATTN_DIM_C * sizeof(float) + 255) & ~(size_t)255;
    float* qr_vec   = (float*)ws;                                  // 64
    ws += 256;
    float* agg      = (float*)ws;                                  // 50000*128
    (void)ws_size; (void)n_in; (void)in_sizes; (void)out_size;

    const int mTiles = N_NODE_C / 16;                              // 3125
    const int gemmBlocks = (mTiles + 7) / 8;                       // 391

    // 1) agg = 0
    {
        const int n = N_NODE_C * IN_DIM_C;
        gnn_zero_kernel<<<(n + 255) / 256, 256, 0, stream>>>(agg, n);
    }
    // 2) qr_vec = rela[q_rel[r_idx]] @ Wqr_w + Wqr_b
    gnn_qr_kernel<<<1, ATTN_DIM_C, 0, stream>>>(q_rel, r_idx, rela, Wqr_w, Wqr_b, qr_vec);
    // 3) att_rel = rela @ Wr + qr_vec
    gnn_relatt_kernel<<<N_REL_C, ATTN_DIM_C, 0, stream>>>(rela, Wr, qr_vec, att_rel);
    // 4) att_node = hidden @ Ws   (50000x128 @ 128x64, WMMA f32, LDS-staged B)
    gnn_wmma_gemm_f32<ATTN_DIM_C, 4><<<gemmBlocks, 256, 0, stream>>>(
        hidden, Ws, att_node, N_NODE_C);
    // 5) edge stage: alpha + scatter-add messages
    {
        const int wavesPerBlock = 8;                               // 256 threads
        const int blocks = (N_EDGE_C + wavesPerBlock - 1) / wavesPerBlock;
        gnn_edge_kernel<<<blocks, 32 * wavesPerBlock, 0, stream>>>(
            edges, att_node, att_rel, hidden, rela, wa_w, wa_b, agg);
    }
    // 6) out = agg @ Wh   (50000x128 @ 128x128, WMMA f32, LDS-staged B)
    gnn_wmma_gemm_f32<OUT_DIM_C, 8><<<gemmBlocks, 256, 0, stream>>>(
        agg, Wh, out, N_NODE_C);
}